// PointSampler_55808805044338
// MI455X (gfx1250) — compile-verified
//
#include <hip/hip_runtime.h>
#include <hip/hip_bf16.h>

typedef __attribute__((ext_vector_type(8))) float v8f;
typedef __attribute__((ext_vector_type(2))) float v2f;

#define NB 16
#define NV 100000
#define NF 200000
#define NP 65536

__device__ __forceinline__ int clampi(int x, int lo, int hi) {
    return x < lo ? lo : (x > hi ? hi : x);
}

// ---------------------------------------------------------------------------
// Kernel 1: per-face area  ->  ws[b*NF + f]
// ---------------------------------------------------------------------------
__global__ __launch_bounds__(256) void area_kernel(
    const float* __restrict__ verts, const int* __restrict__ faces,
    float* __restrict__ areas) {
    int gid = blockIdx.x * blockDim.x + threadIdx.x;
    if (gid >= NB * NF) return;
    int b = gid / NF;
    const int* fp = faces + (size_t)gid * 3;
    const float* vb = verts + (size_t)b * NV * 3;
    int i0 = clampi(fp[0], 0, NV - 1);
    int i1 = clampi(fp[1], 0, NV - 1);
    int i2 = clampi(fp[2], 0, NV - 1);
    float ax = vb[3*i0+0], ay = vb[3*i0+1], az = vb[3*i0+2];
    float bx = vb[3*i1+0], by = vb[3*i1+1], bz = vb[3*i1+2];
    float cx = vb[3*i2+0], cy = vb[3*i2+1], cz = vb[3*i2+2];
    float e1x = bx - ax, e1y = by - ay, e1z = bz - az;
    float e2x = cx - ax, e2y = cy - ay, e2z = cz - az;
    float nx = e1y * e2z - e1z * e2y;
    float ny = e1z * e2x - e1x * e2z;
    float nz = e1x * e2y - e1y * e2x;
    areas[gid] = 0.5f * sqrtf(nx * nx + ny * ny + nz * nz);
}

// ---------------------------------------------------------------------------
// Kernel 2: in-place inclusive scan of ws[b*NF .. b*NF+NF) per batch.
// One block (8 waves) per batch. Each wave scans a 256-element tile as a
// 16x16 matrix using V_WMMA_F32_16X16X4_F32:
//   chain 1:  R  = X * U          (U upper-triangular ones -> row prefix sums)
//   chain 2:  D  = R + Lstrict*T  (T = rowTot broadcast -> cross-row offsets)
// Wave totals are combined across the 8 waves in LDS; a per-thread replicated
// running carry links the 2048-element super-chunks.
// ---------------------------------------------------------------------------
__global__ __launch_bounds__(256) void scan_kernel(float* __restrict__ data) {
    __shared__ float xs[2048];
    __shared__ float rowtot[8][16];
    __shared__ float wt[8];

    const int tid  = threadIdx.x;
    const int wave = tid >> 5;
    const int lane = tid & 31;
    const int half = lane >> 4;     // K-half of the wave (A/B f32 layout)
    const int ln   = lane & 15;     // M for A, N for B/C/D
    const int kb   = half * 2;

    float* g = data + (size_t)blockIdx.x * NF;
    float carry = 0.0f;
    const int nchunks = (NF + 2047) / 2048;

    for (int c = 0; c < nchunks; ++c) {
        const int base = c * 2048;
        // coalesced cooperative load (zero-pad the tail)
        #pragma unroll
        for (int t = 0; t < 8; ++t) {
            int off = tid + 256 * t;
            int e = base + off;
            xs[off] = (e < NF) ? g[e] : 0.0f;
        }
        __syncthreads();

        const float* xw = xs + wave * 256;   // this wave's 16x16 tile X
        v8f acc = {};

        // chain 1: acc = X * U   (A = X 16x4 slices, B = triangular mask)
        #pragma unroll
        for (int k = 0; k < 4; ++k) {
            const int kk = 4 * k + kb;
            v2f a, bb;
            a.x  = xw[16 * ln + kk];
            a.y  = xw[16 * ln + kk + 1];
            bb.x = (kk     <= ln) ? 1.0f : 0.0f;
            bb.y = (kk + 1 <= ln) ? 1.0f : 0.0f;
            acc = __builtin_amdgcn_wmma_f32_16x16x4_f32(
                false, a, false, bb, (short)0, acc, false, false);
        }

        // row totals = column 15 of R (held by lanes 15 and 31)
        if (ln == 15) {
            #pragma unroll
            for (int r = 0; r < 8; ++r) rowtot[wave][r + 8 * half] = acc[r];
        }
        __syncthreads();

        // chain 2: acc += Lstrict * broadcast(rowTot)
        #pragma unroll
        for (int k = 0; k < 4; ++k) {
            const int kk = 4 * k + kb;
            v2f a, bb;
            a.x  = (kk     < ln) ? 1.0f : 0.0f;
            a.y  = (kk + 1 < ln) ? 1.0f : 0.0f;
            bb.x = rowtot[wave][kk];
            bb.y = rowtot[wave][kk + 1];
            acc = __builtin_amdgcn_wmma_f32_16x16x4_f32(
                false, a, false, bb, (short)0, acc, false, false);
        }

        // wave total = scanned tile's last element = lane31's acc[7]
        if (lane == 31) wt[wave] = acc[7];
        __syncthreads();

        float prefix = carry;
        #pragma unroll
        for (int w = 0; w < 8; ++w) {
            float t = wt[w];
            if (w < wave) prefix += t;
            carry += t;                 // identical in every thread
        }

        // store scanned values (C/D layout: M = r + 8*half, N = ln)
        #pragma unroll
        for (int r = 0; r < 8; ++r) {
            int e = base + wave * 256 + 16 * (r + 8 * half) + ln;
            if (e < NF) g[e] = acc[r] + prefix;
        }
        __syncthreads();
    }
}

// ---------------------------------------------------------------------------
// Kernel 3: inverse-CDF sample + barycentric blend. One thread per (b,p).
// ---------------------------------------------------------------------------
__global__ __launch_bounds__(256) void sample_kernel(
    const float* __restrict__ verts, const int* __restrict__ faces,
    const float* __restrict__ eps1, const float* __restrict__ eps2,
    const float* __restrict__ uu, const float* __restrict__ cdf,
    float* __restrict__ out) {
    int gid = blockIdx.x * blockDim.x + threadIdx.x;
    if (gid >= NB * NP) return;
    int b = gid / NP;

    const float* c = cdf + (size_t)b * NF;
    float total  = c[NF - 1];
    float target = uu[gid] * total;   // searchsorted(cdf/total, u) == first cdf >= u*total

    int lo = 0, hi = NF;
    while (lo < hi) {
        int mid = (lo + hi) >> 1;
        if (c[mid] < target) lo = mid + 1; else hi = mid;
    }
    int idx = clampi(lo, 0, NF - 1);

    const int* fp = faces + ((size_t)b * NF + idx) * 3;
    const float* vb = verts + (size_t)b * NV * 3;
    int i0 = clampi(fp[0], 0, NV - 1);
    int i1 = clampi(fp[1], 0, NV - 1);
    int i2 = clampi(fp[2], 0, NV - 1);
    float ax = vb[3*i0+0], ay = vb[3*i0+1], az = vb[3*i0+2];
    float bx = vb[3*i1+0], by = vb[3*i1+1], bz = vb[3*i1+2];
    float cx = vb[3*i2+0], cy = vb[3*i2+1], cz = vb[3*i2+2];

    float e1x = bx - ax, e1y = by - ay, e1z = bz - az;
    float e2x = cx - ax, e2y = cy - ay, e2z = cz - az;
    float nx = e1y * e2z - e1z * e2y;
    float ny = e1z * e2x - e1x * e2z;
    float nz = e1x * e2y - e1y * e2x;
    float nn = sqrtf(nx * nx + ny * ny + nz * nz);

    float s  = sqrtf(eps1[gid]);
    float e2v = eps2[gid];
    float w0 = 1.0f - s;
    float w1 = (1.0f - e2v) * s;
    float w2 = e2v * s;

    size_t o = (size_t)gid * 3;
    out[o + 0] = w0 * ax + w1 * bx + w2 * cx;
    out[o + 1] = w0 * ay + w1 * by + w2 * cy;
    out[o + 2] = w0 * az + w1 * bz + w2 * cz;

    size_t o2 = (size_t)NB * NP * 3 + o;
    out[o2 + 0] = nx / nn;
    out[o2 + 1] = ny / nn;
    out[o2 + 2] = nz / nn;
}

// ---------------------------------------------------------------------------
extern "C" void kernel_launch(void* const* d_in, const int* in_sizes, int n_in,
                              void* d_out, int out_size, void* d_ws, size_t ws_size,
                              hipStream_t stream) {
    const float* verts = (const float*)d_in[0];
    const int*   faces = (const int*)d_in[1];
    const float* e1    = (const float*)d_in[2];
    const float* e2    = (const float*)d_in[3];
    const float* u     = (const float*)d_in[4];
    float* cdf = (float*)d_ws;            // NB*NF floats = 12.8 MB
    float* out = (float*)d_out;

    int nfaces = NB * NF;
    area_kernel<<<(nfaces + 255) / 256, 256, 0, stream>>>(verts, faces, cdf);
    scan_kernel<<<NB, 256, 0, stream>>>(cdf);
    int npts = NB * NP;
    sample_kernel<<<(npts + 255) / 256, 256, 0, stream>>>(verts, faces, e1, e2, u, cdf, out);
}